// GAT_35296041238878
// MI455X (gfx1250) — compile-verified
//
#include <hip/hip_runtime.h>
#include <hip/hip_bf16.h>

typedef float v2f __attribute__((ext_vector_type(2)));
typedef float v8f __attribute__((ext_vector_type(8)));

#define NEG_SLOPE 0.2f

// ---------------------------------------------------------------------------
// fp32 WMMA GEMM: C[M,Ncols] = A[M,K] @ B[K,Ncols]
// block = (Ncols/16) waves * 32; blockIdx.x = M-tile (16 rows).
// A tile staged in LDS (padded), each wave owns one 16-wide N tile.
// Uses V_WMMA_F32_16X16X4_F32 (full fp32 precision, no downconvert).
// ---------------------------------------------------------------------------
__global__ __launch_bounds__(256) void gat_gemm_wmma(
    const float* __restrict__ A, const float* __restrict__ B,
    float* __restrict__ C, int M, int K, int Ncols)
{
    extern __shared__ float ldsA[];      // 16 * (K + 4) floats, padded stride
    const int pK = K + 4;
    const int tid = threadIdx.x;
    const int m0  = blockIdx.x * 16;

    // cooperative A-tile load (zero-fill rows past M)
    for (int i = tid; i < 16 * K; i += blockDim.x) {
        int r = i / K, c = i - r * K;
        int row = m0 + r;
        ldsA[r * pK + c] = (row < M) ? A[(long)row * K + c] : 0.0f;
    }
    __syncthreads();

    const int wave = tid >> 5;
    const int lane = tid & 31;
    const int half = lane >> 4;          // K sub-pair select per ISA A layout
    const int lm   = lane & 15;
    const int n0   = wave * 16;

    v8f acc = {0.f, 0.f, 0.f, 0.f, 0.f, 0.f, 0.f, 0.f};

    for (int k0 = 0; k0 < K; k0 += 4) {
        const int ka = k0 + 2 * half;
        // A frag: lane (lm, half) holds A[m0+lm][ka], A[m0+lm][ka+1]
        v2f av = { ldsA[lm * pK + ka], ldsA[lm * pK + ka + 1] };
        // B frag: lane (lm, half) holds B[ka][n0+lm], B[ka+1][n0+lm]
        v2f bv = { B[(long)ka * Ncols + n0 + lm],
                   B[(long)(ka + 1) * Ncols + n0 + lm] };
        acc = __builtin_amdgcn_wmma_f32_16x16x4_f32(
                  false, av, false, bv, (short)0, acc, false, false);
    }

    // D layout: VGPR j -> row m0 + j + 8*half, col n0 + lm
#pragma unroll
    for (int j = 0; j < 8; j++) {
        int row = m0 + j + 8 * half;
        if (row < M) C[(long)row * Ncols + n0 + lm] = acc[j];
    }
}

// ---------------------------------------------------------------------------
// alpha[n][h] = sum_c h[n][h][c] * a[h][c]   (for both a_src and a_dst)
// ---------------------------------------------------------------------------
__global__ void gat_alpha(const float* __restrict__ h,
                          const float* __restrict__ a_src,
                          const float* __restrict__ a_dst,
                          float* __restrict__ al_src,
                          float* __restrict__ al_dst,
                          int Nn, int H, int Cc)
{
    int idx = blockIdx.x * blockDim.x + threadIdx.x;
    if (idx >= Nn * H) return;
    int n = idx / H, hh = idx - n * H;
    const float* hp = h + (long)n * H * Cc + hh * Cc;
    const float* as = a_src + hh * Cc;
    const float* ad = a_dst + hh * Cc;
    float s = 0.f, d = 0.f;
    for (int c = 0; c < Cc; c++) { float v = hp[c]; s += v * as[c]; d += v * ad[c]; }
    al_src[idx] = s;
    al_dst[idx] = d;
}

__device__ __forceinline__ void atomicMaxFloat(float* addr, float v)
{
    if (v >= 0.f) atomicMax((int*)addr, __float_as_int(v));
    else          atomicMin((unsigned int*)addr, __float_as_uint(v));
}

__device__ __forceinline__ void edge_nodes(const int* __restrict__ ei, int e,
                                           int E_, int& s, int& d)
{
    if (e < E_) { s = ei[e]; d = ei[E_ + e]; }
    else        { s = e - E_; d = s; }       // virtual self-loop
}

// pass 1: segment max of leaky_relu logits over destinations
__global__ void gat_edge_max(const int* __restrict__ ei, int E_, int Etot,
                             const float* __restrict__ als,
                             const float* __restrict__ ald,
                             float* __restrict__ emax, int H)
{
    int e = blockIdx.x * blockDim.x + threadIdx.x;
    if (e >= Etot) return;
    int s, d; edge_nodes(ei, e, E_, s, d);
    for (int hh = 0; hh < H; hh++) {
        float v = als[s * H + hh] + ald[d * H + hh];
        v = (v > 0.f) ? v : NEG_SLOPE * v;
        atomicMaxFloat(&emax[d * H + hh], v);
    }
}

// pass 2: exp(e - max) per edge + segment-sum denominator
__global__ void gat_edge_exp(const int* __restrict__ ei, int E_, int Etot,
                             const float* __restrict__ als,
                             const float* __restrict__ ald,
                             const float* __restrict__ emax,
                             float* __restrict__ ex,
                             float* __restrict__ denom, int H)
{
    int e = blockIdx.x * blockDim.x + threadIdx.x;
    if (e >= Etot) return;
    int s, d; edge_nodes(ei, e, E_, s, d);
    for (int hh = 0; hh < H; hh++) {
        float v = als[s * H + hh] + ald[d * H + hh];
        v = (v > 0.f) ? v : NEG_SLOPE * v;
        float ev = expf(v - emax[d * H + hh]);
        ex[(long)e * H + hh] = ev;
        atomicAdd(&denom[d * H + hh], ev);
    }
}

// pass 3: weighted scatter-add of source features into destinations
__global__ void gat_aggregate(const int* __restrict__ ei, int E_, int Etot,
                              const float* __restrict__ ex,
                              const float* __restrict__ denom,
                              const float* __restrict__ h,
                              float* __restrict__ acc, int H, int Cc)
{
    long gid = (long)blockIdx.x * blockDim.x + threadIdx.x;
    const int F = H * Cc;
    if (gid >= (long)Etot * F) return;
    int e = (int)(gid / F);
    int f = (int)(gid - (long)e * F);
    int hh = f / Cc;
    int s, d; edge_nodes(ei, e, E_, s, d);
    float att = ex[(long)e * H + hh] / (denom[d * H + hh] + 1e-16f);
    atomicAdd(&acc[(long)d * F + f], att * h[(long)s * F + f]);
}

__global__ void gat_bias_elu(const float* __restrict__ acc,
                             const float* __restrict__ b,
                             float* __restrict__ outp, int Nn, int F)
{
    long gid = (long)blockIdx.x * blockDim.x + threadIdx.x;
    if (gid >= (long)Nn * F) return;
    int f = (int)(gid % F);
    float v = acc[gid] + b[f];
    outp[gid] = (v > 0.f) ? v : (expf(v) - 1.0f);
}

__global__ void gat_bias(const float* __restrict__ acc,
                         const float* __restrict__ b,
                         float* __restrict__ outp, int Nn, int F)
{
    long gid = (long)blockIdx.x * blockDim.x + threadIdx.x;
    if (gid >= (long)Nn * F) return;
    int f = (int)(gid % F);
    outp[gid] = acc[gid] + b[f];
}

__global__ void gat_fill(float* __restrict__ p, float v, long n)
{
    long gid = (long)blockIdx.x * blockDim.x + threadIdx.x;
    if (gid < n) p[gid] = v;
}

// ---------------------------------------------------------------------------
extern "C" void kernel_launch(void* const* d_in, const int* in_sizes, int n_in,
                              void* d_out, int out_size, void* d_ws, size_t ws_size,
                              hipStream_t stream)
{
    const float* x   = (const float*)d_in[0];
    const int*   ei  = (const int*)  d_in[1];   // [2, E] int32
    const float* W1  = (const float*)d_in[2];
    const float* a1s = (const float*)d_in[3];
    const float* a1d = (const float*)d_in[4];
    const float* b1  = (const float*)d_in[5];
    const float* W2  = (const float*)d_in[6];
    const float* a2s = (const float*)d_in[7];
    const float* a2d = (const float*)d_in[8];
    const float* b2  = (const float*)d_in[9];
    float* out = (float*)d_out;

    const int Nn   = 50000;
    const int E_   = in_sizes[1] / 2;           // 800000
    const int Etot = E_ + Nn;                   // + self loops
    const int K    = 128;                       // IN_DIM == HEADS*HID
    const int H1 = 4, C1 = 32, F1 = 128;
    const int H2 = 1, C2 = 32;

    // workspace carve-up (floats)
    float* ws    = (float*)d_ws;
    float* h1    = ws; ws += (long)Nn * F1;
    float* acc1  = ws; ws += (long)Nn * F1;
    float* al1s  = ws; ws += Nn * H1;
    float* al1d  = ws; ws += Nn * H1;
    float* emax1 = ws; ws += Nn * H1;
    float* den1  = ws; ws += Nn * H1;
    float* ex1   = ws; ws += (long)Etot * H1;
    float* h2    = ws; ws += (long)Nn * C2;
    float* acc2  = ws; ws += (long)Nn * C2;
    float* al2s  = ws; ws += Nn;
    float* al2d  = ws; ws += Nn;
    float* emax2 = ws; ws += Nn;
    float* den2  = ws; ws += Nn;
    float* ex2   = ws; ws += Etot;

    const int TB = 256;
    auto blocks = [](long n, int tb) { return (unsigned)((n + tb - 1) / tb); };

    // -------- init accumulators (every call; harness does not re-poison) ----
    gat_fill<<<blocks((long)Nn * F1, TB), TB, 0, stream>>>(acc1, 0.f, (long)Nn * F1);
    gat_fill<<<blocks(Nn * H1, TB), TB, 0, stream>>>(emax1, -INFINITY, Nn * H1);
    gat_fill<<<blocks(Nn * H1, TB), TB, 0, stream>>>(den1, 0.f, Nn * H1);
    gat_fill<<<blocks((long)Nn * C2, TB), TB, 0, stream>>>(acc2, 0.f, (long)Nn * C2);
    gat_fill<<<blocks(Nn, TB), TB, 0, stream>>>(emax2, -INFINITY, Nn);
    gat_fill<<<blocks(Nn, TB), TB, 0, stream>>>(den2, 0.f, Nn);

    const int mtiles = (Nn + 15) / 16;
    const size_t lds_bytes = 16 * (K + 4) * sizeof(float);

    // -------- layer 1 ------------------------------------------------------
    // h1 = x @ W1  (fp32 WMMA), 8 waves/block (Ncols=128)
    gat_gemm_wmma<<<dim3(mtiles, 1, 1), (F1 / 16) * 32, lds_bytes, stream>>>(
        x, W1, h1, Nn, K, F1);
    gat_alpha<<<blocks(Nn * H1, TB), TB, 0, stream>>>(h1, a1s, a1d, al1s, al1d, Nn, H1, C1);
    gat_edge_max<<<blocks(Etot, TB), TB, 0, stream>>>(ei, E_, Etot, al1s, al1d, emax1, H1);
    gat_edge_exp<<<blocks(Etot, TB), TB, 0, stream>>>(ei, E_, Etot, al1s, al1d, emax1, ex1, den1, H1);
    gat_aggregate<<<blocks((long)Etot * F1, TB), TB, 0, stream>>>(
        ei, E_, Etot, ex1, den1, h1, acc1, H1, C1);
    // elu(acc1 + b1) -> reuse h1 buffer as layer-2 input
    gat_bias_elu<<<blocks((long)Nn * F1, TB), TB, 0, stream>>>(acc1, b1, h1, Nn, F1);

    // -------- layer 2 ------------------------------------------------------
    // h2 = h1 @ W2  (fp32 WMMA), 2 waves/block (Ncols=32)
    gat_gemm_wmma<<<dim3(mtiles, 1, 1), (C2 / 16) * 32, lds_bytes, stream>>>(
        h1, W2, h2, Nn, K, C2);
    gat_alpha<<<blocks(Nn * H2, TB), TB, 0, stream>>>(h2, a2s, a2d, al2s, al2d, Nn, H2, C2);
    gat_edge_max<<<blocks(Etot, TB), TB, 0, stream>>>(ei, E_, Etot, al2s, al2d, emax2, H2);
    gat_edge_exp<<<blocks(Etot, TB), TB, 0, stream>>>(ei, E_, Etot, al2s, al2d, emax2, ex2, den2, H2);
    gat_aggregate<<<blocks((long)Etot * C2, TB), TB, 0, stream>>>(
        ei, E_, Etot, ex2, den2, h2, acc2, H2, C2);
    gat_bias<<<blocks((long)Nn * C2, TB), TB, 0, stream>>>(acc2, b2, out, Nn, C2);
}